// MetricLoss_81252191306269
// MI455X (gfx1250) — compile-verified
//
#include <hip/hip_runtime.h>
#include <math.h>

#define B_   8
#define P_   4096
#define N_   (B_*P_)
#define D_   32
#define K_   36
#define KNB  (K_-1)
#define SEG  (P_/32)          // 128 candidates per lane segment

static __device__ __forceinline__ float kl_gauss(float sum_mu2, float var) {
    const float VP = 1.0f / 96.0f;
    const float LOG_VP = -4.5643481914678361f; // log(1/96)
    return 0.5f * (var / VP * (float)D_ + sum_mu2 / VP - (float)D_ +
                   (float)D_ * (LOG_VP - logf(var)));
}

typedef float v2f __attribute__((ext_vector_type(2)));
typedef float v8f __attribute__((ext_vector_type(8)));

// order-preserving float -> uint mapping (total order, NaN-free data here)
static __device__ __forceinline__ unsigned flip_f32(float f) {
    unsigned u = __float_as_uint(f);
    return u ^ ((u & 0x80000000u) ? 0xFFFFFFFFu : 0x80000000u);
}

// ---------------------------------------------------------------------------
// K0: pack WMMA operands. q4 = (x,y,z,1), c4 = (-2x,-2y,-2z,|q|^2).
//     Also zero the 8 float accumulators (must happen every launch).
// ---------------------------------------------------------------------------
__global__ void prep_kernel(const float* __restrict__ xyz,
                            float* __restrict__ q4, float* __restrict__ c4,
                            float* __restrict__ acc) {
    int i = blockIdx.x * blockDim.x + threadIdx.x;
    if (i < 8) acc[i] = 0.0f;
    if (i >= N_) return;
    float x = xyz[4*i+1], y = xyz[4*i+2], z = xyz[4*i+3];
    q4[4*i+0] = x;        q4[4*i+1] = y;
    q4[4*i+2] = z;        q4[4*i+3] = 1.0f;
    c4[4*i+0] = -2.0f*x;  c4[4*i+1] = -2.0f*y;
    c4[4*i+2] = -2.0f*z;  c4[4*i+3] = x*x + y*y + z*z;
}

// ---------------------------------------------------------------------------
// K1: KNN. One 256-thread block per (batch, 16-query block).
//   Phase A: 8 waves tile the 4096 candidate columns with
//            V_WMMA_F32_16X16X4_F32 (d_eff = |q|^2 - 2 p.q; row-constant
//            |p|^2 dropped since per-row ordering is unchanged), results
//            flipped to orderable u32 keys and stored to a 16x4096 LDS slab.
//   Phase B: each wave owns 2 rows. Per row: lane L caches the min key of
//            its 128-column segment; 36 extractions each do a 5-step wave32
//            butterfly over the 32 cached minima, knock out the winner in
//            LDS, then cooperatively rescan ONLY the winner's segment
//            (4 elems/lane) to refresh its cached minimum. Sorted order and
//            lowest-index tie-break match jax top_k; extraction 0 = self.
// ---------------------------------------------------------------------------
__launch_bounds__(256)
__global__ void knn_kernel(const float* __restrict__ q4,
                           const float* __restrict__ c4,
                           int* __restrict__ nb) {
    extern __shared__ unsigned dist[];           // [16][P_] u32 keys = 256 KB
    const int blk   = blockIdx.x;
    const int b     = blk / (P_/16);
    const int row0  = (blk % (P_/16)) * 16;
    const int gbase = b * P_ + row0;
    const int lane  = threadIdx.x & 31;
    const int wave  = threadIdx.x >> 5;
    const int half  = lane >> 4;                 // 0: K=0,1   1: K=2,3 (A/B layout)
    const int l16   = lane & 15;

    // A fragment (16x4 f32): lane holds two consecutive K components of row l16.
    const float* ap = q4 + 4*(size_t)(gbase + l16) + 2*half;
    v2f a; a.x = ap[0]; a.y = ap[1];

    // ---- Phase A: distance tiles via WMMA, flipped keys into LDS ----
    for (int t = wave; t < P_/16; t += 8) {
        const int col = t * 16;
        const float* bp = c4 + 4*((size_t)b*P_ + col + l16) + 2*half;
        if (t + 8 < P_/16)
            __builtin_prefetch(c4 + 4*((size_t)b*P_ + (t+8)*16 + l16) + 2*half, 0, 3);
        v2f bb; bb.x = bp[0]; bb.y = bp[1];
        v8f c = {};
        c = __builtin_amdgcn_wmma_f32_16x16x4_f32(
                /*neg_a=*/false, a, /*neg_b=*/false, bb,
                /*c_mod=*/(short)0, c, /*reuse_a=*/false, /*reuse_b=*/false);
        #pragma unroll
        for (int r = 0; r < 8; ++r) {
            // D layout: VGPR r -> M = r (lanes 0-15) or r+8 (lanes 16-31), N = l16
            dist[(r + 8*half) * P_ + col + l16] = flip_f32(c[r]);
        }
    }
    __syncthreads();

    // ---- Phase B: segmented top-36 extraction ----
    for (int rr = 0; rr < 2; ++rr) {
        const int row = wave + 8*rr;
        unsigned* drow = dist + (size_t)row * P_;

        // build per-lane segment minimum (key = dist_u32 <<32 | col)
        const int seg = lane * SEG;
        unsigned long long lmin = ~0ull;
        #pragma unroll 8
        for (int j = 0; j < SEG; ++j) {
            const int c0 = seg + j;
            unsigned long long key = ((unsigned long long)drow[c0] << 32) | (unsigned)c0;
            lmin = lmin < key ? lmin : key;
        }

        for (int iter = 0; iter < K_; ++iter) {
            // global min over 32 cached segment minima
            unsigned long long best = lmin;
            #pragma unroll
            for (int off = 16; off > 0; off >>= 1) {
                unsigned long long o = __shfl_xor(best, off, 32);
                best = best < o ? best : o;
            }
            const int idx    = (int)(best & 0xFFFFFFFFu);
            const int winner = idx >> 7;                   // idx / SEG

            if (lane == 0 && iter > 0)
                nb[(size_t)(gbase + row) * KNB + (iter - 1)] = b * P_ + idx;

            // knock out extracted element (one lane), ensure visible in-wave
            if (lane == (idx & 31))
                drow[idx] = 0xFFFFFFFFu;
            asm volatile("s_wait_dscnt 0" ::: "memory");

            // cooperatively rescan ONLY the winner's segment: 4 elems/lane
            const int base = idx & ~(SEG - 1);
            unsigned long long nm = ~0ull;
            #pragma unroll
            for (int j = 0; j < 4; ++j) {
                const int c0 = base + lane + 32*j;
                unsigned long long key = ((unsigned long long)drow[c0] << 32) | (unsigned)c0;
                nm = nm < key ? nm : key;
            }
            #pragma unroll
            for (int off = 16; off > 0; off >>= 1) {
                unsigned long long o = __shfl_xor(nm, off, 32);
                nm = nm < o ? nm : o;
            }
            if (lane == winner) lmin = nm;
        }
    }
}

// ---------------------------------------------------------------------------
// K2: per-point triplet loss. Gathers only the 2 needed neighbor features.
//     Accumulates w-weighted sums with float global atomics.
// ---------------------------------------------------------------------------
__global__ void loss_kernel(const float* __restrict__ feature,
                            const float* __restrict__ sigma,
                            const int*   __restrict__ label,
                            const int*   __restrict__ pos_idx,
                            const int*   __restrict__ neg_idx,
                            const int*   __restrict__ nb,
                            float* __restrict__ acc) {
    int i = blockIdx.x * blockDim.x + threadIdx.x;
    if (i >= N_) return;
    const int* nbi = nb + (size_t)i * KNB;
    const int lab = label[i];

    unsigned long long mask = 0ull;
    int cnt = 0;
    #pragma unroll 5
    for (int j = 0; j < KNB; ++j) {
        int m = (label[nbi[j]] == lab);
        cnt += m;
        mask |= ((unsigned long long)m) << j;
    }
    const int pi = pos_idx[i], ni = neg_idx[i];
    const bool pos_real = (mask >> pi) & 1ull;
    const bool neg_real = !((mask >> ni) & 1ull);
    const bool point_mask = (cnt > 0) && (cnt < KNB);
    if (!(point_mask && pos_real && neg_real)) return;   // w == 0 contributes nothing

    const int p = nbi[pi], n = nbi[ni];
    const float vA = sigma[i], vP = sigma[p], vN = sigma[n];

    const float* fA = feature + (size_t)i * D_;
    const float* fP = feature + (size_t)p * D_;
    const float* fN = feature + (size_t)n * D_;

    float na = 0.f, np = 0.f, nn = 0.f;
    #pragma unroll
    for (int d = 0; d < D_; ++d) {
        na = fmaf(fA[d], fA[d], na);
        np = fmaf(fP[d], fP[d], np);
        nn = fmaf(fN[d], fN[d], nn);
    }
    const float sA = rsqrtf(na), sP = rsqrtf(np), sN = rsqrtf(nn);

    float mu = 0.f, s2 = 0.f, sumA2 = 0.f, sumP2 = 0.f, sumN2 = 0.f;
    #pragma unroll
    for (int d = 0; d < D_; ++d) {
        const float a  = fA[d] * sA;
        const float pp = fP[d] * sP;
        const float qq = fN[d] * sN;
        const float a2 = a*a, p2 = pp*pp, n2 = qq*qq;
        sumA2 += a2; sumP2 += p2; sumN2 += n2;
        mu += p2 - n2 - 2.0f * a * (pp - qq);
        const float T1 = vP*vP + 2.f*p2*vP + 2.f*(vA + a2)*(vP + p2) - 2.f*a2*p2 - 4.f*a*pp*vP;
        const float T2 = vN*vN + 2.f*n2*vN + 2.f*(vA + a2)*(vN + n2) - 2.f*a2*n2 - 4.f*a*qq*vN;
        const float T3 = 4.f*pp*qq*vA;
        s2 += 2.f*T1 + 2.f*T2 - 2.f*T3;
    }
    mu += (float)D_ * (vP - vN);

    const float sig   = sqrtf(s2 + 1e-7f);
    // Phi(0; mu, s) = 0.5 * erfc(mu / (s*sqrt(2)))
    const float probs = 0.5f * erfcf(mu * 0.70710678118654752f / (sig + 1e-8f));
    const float nll   = -logf(probs + 1e-8f);
    const float kls   = kl_gauss(sumA2, vA) + kl_gauss(sumP2, vP) + kl_gauss(sumN2, vN);

    atomicAdd(&acc[0], 1.0f);   // w sum
    atomicAdd(&acc[1], nll);
    atomicAdd(&acc[2], probs);
    atomicAdd(&acc[3], mu);
    atomicAdd(&acc[4], sig);
    atomicAdd(&acc[5], kls);
}

// ---------------------------------------------------------------------------
// K3: finalize the 4 scalar outputs.
// ---------------------------------------------------------------------------
__global__ void final_kernel(const float* __restrict__ acc, float* __restrict__ out) {
    if (blockIdx.x == 0 && threadIdx.x == 0) {
        const float ws = fmaxf(acc[0], 1.0f);
        out[0] = acc[1] / ws + 1e-6f * (acc[5] / ws);  // loss
        out[1] = acc[2] / ws;                          // probs_m
        out[2] = acc[3] / ws;                          // mu_m
        out[3] = acc[4] / ws;                          // sig_m
    }
}

// ---------------------------------------------------------------------------
extern "C" void kernel_launch(void* const* d_in, const int* in_sizes, int n_in,
                              void* d_out, int out_size, void* d_ws, size_t ws_size,
                              hipStream_t stream) {
    const float* feature = (const float*)d_in[0];   // [N, 32]
    const float* sigma   = (const float*)d_in[1];   // [N, 1]
    const float* xyz     = (const float*)d_in[2];   // [N, 4] (bid,x,y,z)
    const int*   label   = (const int*)d_in[3];     // [N, 1]
    const int*   pos_idx = (const int*)d_in[4];     // [N]
    const int*   neg_idx = (const int*)d_in[5];     // [N]
    float* out = (float*)d_out;

    // workspace layout
    float* q4  = (float*)d_ws;                      // N*4 floats
    float* c4  = q4 + (size_t)N_ * 4;               // N*4 floats
    int*   nb  = (int*)(c4 + (size_t)N_ * 4);       // N*35 ints
    float* acc = (float*)(nb + (size_t)N_ * KNB);   // 8 floats

    prep_kernel<<<(N_ + 255) / 256, 256, 0, stream>>>(xyz, q4, c4, acc);

    const size_t lds_bytes = (size_t)16 * P_ * sizeof(unsigned);   // 256 KB (<320 KB/WGP)
    knn_kernel<<<B_ * (P_ / 16), 256, lds_bytes, stream>>>(q4, c4, nb);

    loss_kernel<<<(N_ + 255) / 256, 256, 0, stream>>>(feature, sigma, label,
                                                      pos_idx, neg_idx, nb, acc);
    final_kernel<<<1, 1, 0, stream>>>(acc, out);
}